// VectorQuantizer_38379827757210
// MI455X (gfx1250) — compile-verified
//
#include <hip/hip_runtime.h>
#include <hip/hip_bf16.h>

typedef float v2f __attribute__((ext_vector_type(2)));
typedef float v8f __attribute__((ext_vector_type(8)));

#define DDIM   256
#define KCODES 8192
#define NROWS  32768      // 32*1024
#define MTILE  128        // rows per block (8 waves x 16)
#define NWAVES 8
#define CHUNK  32         // codebook columns per iteration (2 x 16-col WMMA tiles)
#define NCHUNK (KCODES / CHUNK)   // 256
#define GSTEPS (DDIM / 4)         // 64 K4-steps of fp32 WMMA

#define WMMA_F32(a, b, c) \
  __builtin_amdgcn_wmma_f32_16x16x4_f32(false, (a), false, (b), (short)0, (c), false, false)

// ---------------------------------------------------------------------------
// Kernel 1: per-code squared norms  cnorm[k] = sum_d C[d][k]^2
// ---------------------------------------------------------------------------
__global__ __launch_bounds__(256)
void vq_cnorm_kernel(const float* __restrict__ cb, float* __restrict__ cnorm) {
  int k = blockIdx.x * blockDim.x + threadIdx.x;   // coalesced over k
  float s = 0.0f;
#pragma unroll 8
  for (int d = 0; d < DDIM; ++d) {
    float v = cb[d * KCODES + k];
    s += v * v;
  }
  cnorm[k] = s;
}

// ---------------------------------------------------------------------------
// Kernel 2: zero the loss accumulator
// ---------------------------------------------------------------------------
__global__ void vq_zero_kernel(float* acc) {
  if (threadIdx.x == 0 && blockIdx.x == 0) acc[0] = 0.0f;
}

// ---------------------------------------------------------------------------
// Kernel 3: fused distance GEMM (fp32 WMMA 16x16x4) + row argmin
//   score(n,k) = cnorm[k] - 2 * dot(z_n, c_k)   (row norm is argmin-invariant)
//   - 2 column tiles per wave -> A fragment reused twice
//   - even/odd depth split    -> 4 independent WMMA chains per wave
//   - double-buffered B tile  -> global latency hidden under WMMA
// ---------------------------------------------------------------------------
__global__ __launch_bounds__(256)
void vq_argmin_kernel(const float* __restrict__ z,
                      const float* __restrict__ cb,
                      const float* __restrict__ cnorm,
                      int*   __restrict__ idx_out,
                      float* __restrict__ idx_out_f) {
  // Swizzled WMMA staging: element (g, lane, t) holds matrix[d = 4g + 2*(lane/16) + t]
  __shared__ float sA[MTILE * DDIM];       // 128 KB: 8 waves x (16 rows x 256)
  __shared__ float sB[2][CHUNK * DDIM];    // 2 x 32 KB, double buffered
  __shared__ float redV[NWAVES * 8 * 32];  // 8 KB
  __shared__ int   redI[NWAVES * 8 * 32];  // 8 KB

  const int tid  = threadIdx.x;
  const int wave = tid >> 5;
  const int lane = tid & 31;
  const int l16  = lane & 15;
  const int rowBase = blockIdx.x * MTILE;

  // ---- stage A (z rows) into swizzled LDS; global reads fully coalesced ----
  for (int i = 0; i < MTILE; ++i) {
    const int d = tid;                       // 0..255 contiguous per row
    const float v = z[(rowBase + i) * DDIM + d];
    const int w2 = i >> 4, m = i & 15;
    const int g = d >> 2, hh = (d >> 1) & 1, t = d & 1;
    sA[w2 * 4096 + ((g * 32 + hh * 16 + m) << 1) + t] = v;
  }

  // ---- prologue: stage B chunk 0 into buffer 0 ----
  {
    const int dRow = tid >> 3;               // 0..31 (x8 passes)
    const int c4   = (tid & 7) << 2;         // 128B segments per 8 threads
#pragma unroll
    for (int p = 0; p < 8; ++p) {
      const int d = p * 32 + dRow;
      const float4 v = *(const float4*)&cb[d * KCODES + c4];
      const int g = d >> 2, hh = (d >> 1) & 1, t = d & 1;
#pragma unroll
      for (int e = 0; e < 4; ++e) {
        const int c = c4 + e;
        const int tile = c >> 4, cc = c & 15;
        sB[0][tile * 4096 + ((g * 32 + hh * 16 + cc) << 1) + t] =
            ((const float*)&v)[e];
      }
    }
  }
  __syncthreads();

  float bestV[8];
  int   bestI[8];
#pragma unroll
  for (int r = 0; r < 8; ++r) { bestV[r] = 3.4e38f; bestI[r] = 0; }

  const float* aBase = &sA[wave * 4096 + (lane << 1)];

  for (int chunk = 0; chunk < NCHUNK; ++chunk) {
    const int colBase = chunk * CHUNK;
    const int buf = chunk & 1;

    // ---- issue global loads for next chunk (latency hidden under WMMA) ----
    float4 ld[8];
    const int dRow = tid >> 3;
    const int c4   = (tid & 7) << 2;
    if (chunk + 1 < NCHUNK) {
#pragma unroll
      for (int p = 0; p < 8; ++p) {
        const int d = p * 32 + dRow;
        ld[p] = *(const float4*)&cb[d * KCODES + colBase + CHUNK + c4];
      }
      if (chunk + 2 < NCHUNK)
        __builtin_prefetch(&cb[tid * KCODES + colBase + 2 * CHUNK], 0, 1);
    }

    // ---- 16x32 dot tile: 128 WMMAs, 4 independent accumulation chains ----
    const float* bBuf = &sB[buf][lane << 1];
    v8f acc00 = {0.f,0.f,0.f,0.f,0.f,0.f,0.f,0.f};
    v8f acc01 = {0.f,0.f,0.f,0.f,0.f,0.f,0.f,0.f};
    v8f acc10 = {0.f,0.f,0.f,0.f,0.f,0.f,0.f,0.f};
    v8f acc11 = {0.f,0.f,0.f,0.f,0.f,0.f,0.f,0.f};
#pragma unroll
    for (int g = 0; g < GSTEPS; g += 2) {
      v2f a0  = *(const v2f*)(aBase + g * 64);
      v2f b00 = *(const v2f*)(bBuf + g * 64);
      v2f b01 = *(const v2f*)(bBuf + 4096 + g * 64);
      acc00 = WMMA_F32(a0, b00, acc00);
      acc01 = WMMA_F32(a0, b01, acc01);
      v2f a1  = *(const v2f*)(aBase + (g + 1) * 64);
      v2f b10 = *(const v2f*)(bBuf + (g + 1) * 64);
      v2f b11 = *(const v2f*)(bBuf + 4096 + (g + 1) * 64);
      acc10 = WMMA_F32(a1, b10, acc10);
      acc11 = WMMA_F32(a1, b11, acc11);
    }

    // lane holds dot[m][n]: m = r + 8*(lane/16), n = colBase + l16 (+16)
    const float cn0 = cnorm[colBase + l16];
    const float cn1 = cnorm[colBase + 16 + l16];
    const int   c0  = colBase + l16;
    const int   c1  = colBase + 16 + l16;
#pragma unroll
    for (int r = 0; r < 8; ++r) {
      const float v0 = cn0 - 2.0f * (acc00[r] + acc10[r]);
      if (v0 < bestV[r]) { bestV[r] = v0; bestI[r] = c0; }   // cols ascend
      const float v1 = cn1 - 2.0f * (acc01[r] + acc11[r]);
      if (v1 < bestV[r]) { bestV[r] = v1; bestI[r] = c1; }
    }

    // ---- store staged regs into the other buffer ----
    if (chunk + 1 < NCHUNK) {
      float* dst = sB[1 - buf];
#pragma unroll
      for (int p = 0; p < 8; ++p) {
        const int d = p * 32 + dRow;
        const int g = d >> 2, hh = (d >> 1) & 1, t = d & 1;
#pragma unroll
        for (int e = 0; e < 4; ++e) {
          const int c = c4 + e;
          const int tile = c >> 4, cc = c & 15;
          dst[tile * 4096 + ((g * 32 + hh * 16 + cc) << 1) + t] =
              ((const float*)&ld[p])[e];
        }
      }
    }
    __syncthreads();   // s_wait_dscnt + barrier: stores visible, reads done
  }

  // ---- cross-lane argmin: each row's candidates live in 16 lanes ----
#pragma unroll
  for (int r = 0; r < 8; ++r) {
    redV[(wave * 8 + r) * 32 + lane] = bestV[r];
    redI[(wave * 8 + r) * 32 + lane] = bestI[r];
  }
  __syncthreads();

  if (tid < MTILE) {
    const int w2 = tid >> 4, m = tid & 15;
    const int r = m & 7, hh = m >> 3;
    float bv = 3.4e38f; int bi = 0x7fffffff;
    for (int j = 0; j < 16; ++j) {
      const int l2 = hh * 16 + j;
      const float v = redV[(w2 * 8 + r) * 32 + l2];
      const int  ci = redI[(w2 * 8 + r) * 32 + l2];
      if (v < bv || (v == bv && ci < bi)) { bv = v; bi = ci; }
    }
    const int grow = rowBase + tid;
    idx_out[grow]   = bi;
    idx_out_f[grow] = (float)bi;
  }
}

// ---------------------------------------------------------------------------
// Kernel 4: gather z_q + accumulate sum((z_q - z)^2)
// ---------------------------------------------------------------------------
__global__ __launch_bounds__(256)
void vq_gather_kernel(const float* __restrict__ z,
                      const float* __restrict__ cb,
                      const int*   __restrict__ idx,
                      float* __restrict__ zq_out,
                      float* __restrict__ lossAcc) {
  __shared__ float red[8];
  const int row = blockIdx.x;
  const int d   = threadIdx.x;           // 256 threads == 256 dims
  const int k   = idx[row];
  const float c  = cb[d * KCODES + k];   // column gather (L2-resident codebook)
  const float zv = z[row * DDIM + d];
  zq_out[row * DDIM + d] = c;            // z_q_ste == z_q numerically
  float s = (c - zv) * (c - zv);
#pragma unroll
  for (int off = 16; off > 0; off >>= 1) s += __shfl_down(s, off, 32);
  if ((threadIdx.x & 31) == 0) red[threadIdx.x >> 5] = s;
  __syncthreads();
  if (threadIdx.x == 0) {
    float t = 0.0f;
    for (int i = 0; i < 8; ++i) t += red[i];
    atomicAdd(lossAcc, t);
  }
}

// ---------------------------------------------------------------------------
// Kernel 5: finalize loss = (BETA + 1) * mean = 1.25 * sum / (N*D)
// ---------------------------------------------------------------------------
__global__ void vq_finalize_kernel(const float* lossAcc, float* out_loss) {
  if (threadIdx.x == 0 && blockIdx.x == 0)
    out_loss[0] = 1.25f * lossAcc[0] / 8388608.0f;
}

// ---------------------------------------------------------------------------
extern "C" void kernel_launch(void* const* d_in, const int* in_sizes, int n_in,
                              void* d_out, int out_size, void* d_ws, size_t ws_size,
                              hipStream_t stream) {
  const float* z  = (const float*)d_in[0];   // (32,1024,256) f32
  const float* cb = (const float*)d_in[1];   // (256,8192)    f32

  float* out      = (float*)d_out;
  float* out_zq   = out;                      // 8388608
  float* out_idx  = out + 8388608;            // 32768 (as float)
  float* out_loss = out + 8388608 + 32768;    // 1

  char* ws = (char*)d_ws;
  float* cnorm   = (float*)ws;                              // 8192 f32
  int*   idxbuf  = (int*)(ws + 8192 * sizeof(float));       // 32768 i32
  float* lossAcc = (float*)(ws + 8192 * 4 + 32768 * 4);     // 1 f32

  vq_cnorm_kernel  <<<KCODES / 256, 256, 0, stream>>>(cb, cnorm);
  vq_zero_kernel   <<<1, 1, 0, stream>>>(lossAcc);
  vq_argmin_kernel <<<NROWS / MTILE, 256, 0, stream>>>(z, cb, cnorm, idxbuf, out_idx);
  vq_gather_kernel <<<NROWS, 256, 0, stream>>>(z, cb, idxbuf, out_zq, lossAcc);
  vq_finalize_kernel<<<1, 1, 0, stream>>>(lossAcc, out_loss);
}